// SimpleGCNLayer_17721035063937
// MI455X (gfx1250) — compile-verified
//
#include <hip/hip_runtime.h>
#include <hip/hip_bf16.h>

#define N_NODES 100000
#define N_EDGES 1600000
#define D_IN    128
#define D_OUT   64

typedef __attribute__((ext_vector_type(2))) float v2f;
typedef __attribute__((ext_vector_type(8))) float v8f;

// ---------------------------------------------------------------------------
// 1) zero degree counters
__global__ void gcn_zero_deg(unsigned* __restrict__ deg, int n) {
    int i = blockIdx.x * blockDim.x + threadIdx.x;
    if (i < n) deg[i] = 0u;
}

// 2) deg[row[e]] += 1 over the 1.6M edges
__global__ void gcn_count_deg(const long long* __restrict__ rows,
                              unsigned* __restrict__ deg, int E) {
    int e = blockIdx.x * blockDim.x + threadIdx.x;
    if (e < E) atomicAdd(&deg[(int)rows[e]], 1u);
}

// 3) dis[i] = rsqrt(deg[i] + 1)   (+1 = self loop; deg>=1 always afterwards)
__global__ void gcn_make_dis(const unsigned* __restrict__ deg,
                             float* __restrict__ dis, int n) {
    int i = blockIdx.x * blockDim.x + threadIdx.x;
    if (i < n) dis[i] = rsqrtf((float)(deg[i] + 1u));
}

// ---------------------------------------------------------------------------
// 4) xw = x @ W via V_WMMA_F32_16X16X4_F32.
// One wave (32 lanes) per 16-row tile; 4 accumulators cover all 64 columns.
// N_NODES = 16 * 6250 exactly -> no remainder, EXEC stays all-1s.
//
// A (16x4 f32) lane layout (ISA 7.12.2): lanes 0-15 hold K={k,k+1} for M=lane,
// lanes 16-31 hold K={k+2,k+3} for M=lane-16.
// B (4x16 f32): VGPR0 lanes 0-15 = row k, lanes 16-31 = row k+2;
//               VGPR1 lanes 0-15 = row k+1, lanes 16-31 = row k+3.
// C/D (16x16 f32): VGPR r: lanes 0-15 -> M=r, lanes 16-31 -> M=r+8, N=lane&15.
__global__ __launch_bounds__(32) void gcn_gemm_wmma(const float* __restrict__ x,
                                                    const float* __restrict__ W,
                                                    float* __restrict__ xw) {
    const int tile    = blockIdx.x;        // 0..6249 (M tiles)
    const int lane    = threadIdx.x;       // 0..31
    const int l15     = lane & 15;
    const int halfSel = lane >> 4;         // 0 -> K pair {k,k+1}, 1 -> {k+2,k+3}
    const int row0    = tile * 16;

    v8f acc0 = {}, acc1 = {}, acc2 = {}, acc3 = {};

    for (int k = 0; k < D_IN; k += 4) {
        // A fragment (contiguous pair -> 8-byte aligned vector load)
        const float* ap = x + (size_t)(row0 + l15) * D_IN + k + halfSel * 2;
        v2f a = *(const v2f*)ap;

        // B fragments for the 4 column tiles
        const float* bp = W + (size_t)(k + halfSel * 2) * D_OUT + l15;
        v2f b0; b0.x = bp[0];  b0.y = bp[0 + D_OUT];
        v2f b1; b1.x = bp[16]; b1.y = bp[16 + D_OUT];
        v2f b2; b2.x = bp[32]; b2.y = bp[32 + D_OUT];
        v2f b3; b3.x = bp[48]; b3.y = bp[48 + D_OUT];

        acc0 = __builtin_amdgcn_wmma_f32_16x16x4_f32(false, a, false, b0,
                                                     (short)0, acc0, false, false);
        acc1 = __builtin_amdgcn_wmma_f32_16x16x4_f32(false, a, false, b1,
                                                     (short)0, acc1, false, false);
        acc2 = __builtin_amdgcn_wmma_f32_16x16x4_f32(false, a, false, b2,
                                                     (short)0, acc2, false, false);
        acc3 = __builtin_amdgcn_wmma_f32_16x16x4_f32(false, a, false, b3,
                                                     (short)0, acc3, false, false);
    }

    // Store C: VGPR r -> row row0 + r + 8*halfSel, col l15 + 16*tileN
    #pragma unroll
    for (int r = 0; r < 8; ++r) {
        int row = row0 + r + halfSel * 8;
        float* op = xw + (size_t)row * D_OUT + l15;
        op[0]  = acc0[r];
        op[16] = acc1[r];
        op[32] = acc2[r];
        op[48] = acc3[r];
    }
}

// ---------------------------------------------------------------------------
// 5) self-loop term also initializes out: out[i][j] = dis[i]^2 * xw[i][j]
__global__ void gcn_selfloop_init(const float* __restrict__ xw,
                                  const float* __restrict__ dis,
                                  float* __restrict__ out, int n) {
    int i = blockIdx.x * blockDim.x + threadIdx.x;
    if (i < n) {
        int node = i >> 6;                 // / D_OUT
        float d = dis[node];
        out[i] = d * d * xw[i];
    }
}

// 6) edge scatter: one wave per edge, 2 columns per lane, L2-resident atomics.
__global__ void gcn_edge_scatter(const long long* __restrict__ ei,
                                 const float* __restrict__ xw,
                                 const float* __restrict__ dis,
                                 float* __restrict__ out, int E) {
    int t    = blockIdx.x * blockDim.x + threadIdx.x;
    int e    = t >> 5;
    int lane = t & 31;
    if (e >= E) return;

    int r = (int)ei[e];            // same address across the wave -> 1 request
    int c = (int)ei[E + e];
    float s = dis[r] * dis[c];

    const float2* src = (const float2*)(xw + (size_t)c * D_OUT);
    float2 v = src[lane];          // columns 2*lane, 2*lane+1

    float* dst = out + (size_t)r * D_OUT + 2 * lane;
    atomicAdd(dst + 0, s * v.x);
    atomicAdd(dst + 1, s * v.y);
}

// ---------------------------------------------------------------------------
extern "C" void kernel_launch(void* const* d_in, const int* in_sizes, int n_in,
                              void* d_out, int out_size, void* d_ws, size_t ws_size,
                              hipStream_t stream) {
    const float*     x  = (const float*)d_in[0];       // [N, 128]
    const long long* ei = (const long long*)d_in[1];   // [2, E] row-major
    const float*     W  = (const float*)d_in[2];       // [128, 64]
    float*           out = (float*)d_out;              // [N, 64]

    // workspace carve: dis[N] | xw[N*64] | deg[N]  (~26.4 MB total)
    float*    dis = (float*)d_ws;
    float*    xw  = dis + N_NODES;
    unsigned* deg = (unsigned*)(xw + (size_t)N_NODES * D_OUT);

    const int TB = 256;

    gcn_zero_deg<<<(N_NODES + TB - 1) / TB, TB, 0, stream>>>(deg, N_NODES);
    gcn_count_deg<<<(N_EDGES + TB - 1) / TB, TB, 0, stream>>>(ei, deg, N_EDGES);
    gcn_make_dis<<<(N_NODES + TB - 1) / TB, TB, 0, stream>>>(deg, dis, N_NODES);

    gcn_gemm_wmma<<<N_NODES / 16, 32, 0, stream>>>(x, W, xw);

    gcn_selfloop_init<<<((N_NODES * D_OUT) + TB - 1) / TB, TB, 0, stream>>>(
        xw, dis, out, N_NODES * D_OUT);

    long long scatter_threads = (long long)N_EDGES * 32;
    gcn_edge_scatter<<<(unsigned)((scatter_threads + TB - 1) / TB), TB, 0, stream>>>(
        ei, xw, dis, out, N_EDGES);
}